// SimpleNeedleRIN_75892072121072
// MI455X (gfx1250) — compile-verified
//
#include <hip/hip_runtime.h>
#include <hip/hip_bf16.h>

#define D_MODEL 64
#define NUM_SIGNALS 128
#define BATCH 256
#define SEQ 4096
#define LUT_SIZE 4096.0f
#define PHI_F 1.6180339887498949f
#define TWO_PI_F 6.2831853071795865f
#define INV_TWO_PI_F 0.15915494309189535f

typedef float v2f __attribute__((ext_vector_type(2)));
typedef float v8f __attribute__((ext_vector_type(8)));

// ---------------------------------------------------------------------------
// Kernel 1: the nonlinear scan. One thread per (batch, channel) chain.
// grid = BATCH blocks, block = D_MODEL (64) threads = 2 waves.
// Token load is wave-uniform -> compiler emits s_load (scalar path);
// embedding row fetch is lane-coalesced (consecutive d -> consecutive floats).
// Next step's token+row are prefetched while the current trig chain runs.
// ---------------------------------------------------------------------------
__global__ void __launch_bounds__(D_MODEL)
rin_scan_kernel(const int* __restrict__ input_ids,
                const float* __restrict__ emb_table,
                float* __restrict__ h_combined)
{
    const int b = blockIdx.x;      // 0..255
    const int d = threadIdx.x;     // 0..63
    const int* idrow = input_ids + (size_t)b * SEQ;

    float h_r = 0.0f, h_i = 0.0f;
    float tphi = 0.0f;

    // prime the pipeline: load row for t=0
    int tok = idrow[0];
    const float* row = emb_table + (size_t)tok * (2 * D_MODEL);
    float w  = row[d];
    float bb = row[D_MODEL + d];

    for (int t = 0; t < SEQ; ++t) {
        // ---- prefetch t+1 (uniform scalar token load + coalesced row) ----
        const int tn = (t + 1 < SEQ) ? idrow[t + 1] : tok;
        const float* rown = emb_table + (size_t)tn * (2 * D_MODEL);
        const float w_n  = rown[d];
        const float b_n  = rown[D_MODEL + d];

        // ---- recurrence step (critical path) ----
        const float wl     = 1.0f + __builtin_fabsf(w);
        const float inv_wl = __builtin_amdgcn_rcpf(wl);
        const float base   = bb + tphi;
        const float th_r   = __builtin_fmaf(h_r, inv_wl, base);
        const float th_i   = __builtin_fmaf(h_i, inv_wl, base);

        // LUT semantics: idx = floor(fract(theta/2pi)*4096); value = sin/cos(2pi*idx/4096)
        // v_sin_f32/v_cos_f32 compute sin/cos(2pi*x) of a fractional input directly.
        const float fr = __builtin_amdgcn_fractf(th_r * INV_TWO_PI_F);
        const float fi = __builtin_amdgcn_fractf(th_i * INV_TWO_PI_F);
        const float qr = __builtin_floorf(fr * LUT_SIZE) * (1.0f / LUT_SIZE);
        const float qi = __builtin_floorf(fi * LUT_SIZE) * (1.0f / LUT_SIZE);

        const float sr = __builtin_amdgcn_sinf(qr);
        const float cr = __builtin_amdgcn_cosf(qr);
        const float si = __builtin_amdgcn_sinf(qi);
        const float ci = __builtin_amdgcn_cosf(qi);

        h_r = cr * ci - sr * si;   // cos(a+b)
        h_i = cr * si + sr * ci;   // sin(a+b)

        // rotate pipeline registers
        tok = tn; w = w_n; bb = b_n;

        // tphi_t = mod(t*phi, 2pi), maintained incrementally (|phi|<2pi)
        tphi += PHI_F;
        if (tphi >= TWO_PI_F) tphi -= TWO_PI_F;
    }

    // h_combined layout: [B, 128] = concat(h_real, h_imag)
    h_combined[(size_t)b * NUM_SIGNALS + d]           = h_r;
    h_combined[(size_t)b * NUM_SIGNALS + D_MODEL + d] = h_i;
}

// ---------------------------------------------------------------------------
// Kernel 2: out = h_combined @ proj_w^T + proj_b via V_WMMA_F32_16X16X4_F32.
// M=256, N=128, K=128. One wave per 16x16 output tile -> 16*8 = 128 tiles.
// A (16x4 f32): lane = M-row; vgpr pair = {K0,K1} (lanes 0-15) / {K2,K3} (16-31)
// B (4x16 f32): lane = N-col; same K striping.
// C/D (16x16 f32, 8 VGPRs): vgpr r -> row r (+8 for lanes 16-31), lane = col.
// ---------------------------------------------------------------------------
__global__ void __launch_bounds__(32)
rin_proj_wmma_kernel(const float* __restrict__ h_combined,
                     const float* __restrict__ proj_w,
                     const float* __restrict__ proj_b,
                     float* __restrict__ out)
{
    const int tile = blockIdx.x;        // 0..127
    const int mt   = tile & 15;         // 16 tiles along M=256
    const int nt   = tile >> 4;         // 8 tiles along N=128
    const int lane = threadIdx.x;       // 0..31
    const int half = lane >> 4;         // 0 or 1
    const int l    = lane & 15;

    const int m = mt * 16 + l;          // A row for this lane
    const int n = nt * 16 + l;          // B col for this lane

    const float* arow = h_combined + (size_t)m * NUM_SIGNALS + 2 * half;
    const float* brow = proj_w     + (size_t)n * NUM_SIGNALS + 2 * half;

    v8f c = {};
#pragma unroll
    for (int k0 = 0; k0 < NUM_SIGNALS; k0 += 4) {
        v2f a, bm;
        a.x  = arow[k0];  a.y  = arow[k0 + 1];
        bm.x = brow[k0];  bm.y = brow[k0 + 1];
        // 8 args: (neg_a, A, neg_b, B, c_mod, C, reuse_a, reuse_b)
        c = __builtin_amdgcn_wmma_f32_16x16x4_f32(
                /*neg_a=*/false, a, /*neg_b=*/false, bm,
                /*c_mod=*/(short)0, c, /*reuse_a=*/false, /*reuse_b=*/false);
    }

    const float bv = proj_b[n];
#pragma unroll
    for (int r = 0; r < 8; ++r) {
        const int row = mt * 16 + r + 8 * half;
        out[(size_t)row * NUM_SIGNALS + n] = c[r] + bv;
    }
}

// ---------------------------------------------------------------------------
// Host launcher
// ---------------------------------------------------------------------------
extern "C" void kernel_launch(void* const* d_in, const int* in_sizes, int n_in,
                              void* d_out, int out_size, void* d_ws, size_t ws_size,
                              hipStream_t stream)
{
    const int*   input_ids = (const int*)  d_in[0];   // (256, 4096)
    const float* emb_table = (const float*)d_in[1];   // (50257, 128)
    const float* proj_w    = (const float*)d_in[2];   // (128, 128)
    const float* proj_b    = (const float*)d_in[3];   // (128,)
    float*       out       = (float*)d_out;           // (256, 128)

    float* h_combined = (float*)d_ws;                 // 256*128 f32 = 128 KB

    rin_scan_kernel<<<BATCH, D_MODEL, 0, stream>>>(input_ids, emb_table, h_combined);

    const int tiles = (BATCH / 16) * (NUM_SIGNALS / 16);  // 128 waves
    rin_proj_wmma_kernel<<<tiles, 32, 0, stream>>>(h_combined, proj_w, proj_b, out);
}